// FcosRT_8881992368367
// MI455X (gfx1250) — compile-verified
//
#include <hip/hip_runtime.h>
#include <hip/hip_bf16.h>

// ---------------- problem constants ----------------
#define NUM_CLASSES 80
#define N_LVL 3
#define K0 2000
#define K1 2000
#define K2 1600
#define NCAND 5600           // K0+K1+K2
#define SORTN 8192           // pow2 >= NCAND, 64KB LDS of u64
#define NW 175               // NCAND/32 exactly
#define ROWTILES 350         // NCAND/16 exactly
#define NBINS 4096
#define SIDE_CAP 8192
#define CONF_THRESH 0.05f
#define NMS_THRESH 0.6f
#define CLASS_OFFSET 10000.0f

// ---------------- workspace layout (bytes) ----------------
#define OFF_HIST      0u          // 3*4096*4 = 49152
#define OFF_MAINCNT   49152u      // 3*4
#define OFF_SIDECNT   49164u      // 3*4
#define OFF_TBIN      49176u      // 3*4
#define OFF_ABOVE     49188u      // 3*4
#define OFF_RSEL      49200u      // 3*4
#define OFF_REMOVED   49216u      // 175*4 = 700 -> 49916
#define ZERO_WORDS    12480       // zero bytes [0, 49920)
#define OFF_CANDU     49920u      // 5600*4  -> 72320
#define OFF_CANDIDX   72320u      // 5600*4  -> 94720
#define OFF_SIDEU     94720u      // 3*8192*4 -> 193024
#define OFF_SIDEIDX   193024u     // 3*8192*4 -> 291328
#define OFF_KEYS      291328u     // 8192*8  -> 356864
#define OFF_SBOX      356864u     // 5600*4*4 -> 446464
#define OFF_MASK      446464u     // 5600*175*4 -> 4366464

typedef __attribute__((ext_vector_type(2))) float v2f;
typedef __attribute__((ext_vector_type(8))) float v8f;
typedef __attribute__((ext_vector_type(4))) int v4i;

// gfx1250 async global->LDS (ASYNCcnt-tracked), if the toolchain declares it
#if defined(__has_builtin)
#if __has_builtin(__builtin_amdgcn_global_load_async_to_lds_b128)
#define HAVE_ASYNC_LDS 1
#endif
#if __has_builtin(__builtin_amdgcn_s_wait_asynccnt)
#define HAVE_WAIT_ASYNC_BUILTIN 1
#endif
#endif

#define AS_GLOBAL __attribute__((address_space(1)))
#define AS_LDS    __attribute__((address_space(3)))

__device__ __forceinline__ void wait_asynccnt0() {
#if defined(HAVE_WAIT_ASYNC_BUILTIN)
  __builtin_amdgcn_s_wait_asynccnt(0);
#else
  asm volatile("s_wait_asynccnt 0x0" ::: "memory");
#endif
}

// monotonic bijection float-bits -> orderable uint
__device__ __forceinline__ unsigned mapf(float x) {
  unsigned b = __float_as_uint(x);
  return b ^ ((b & 0x80000000u) ? 0xFFFFFFFFu : 0x80000000u);
}
__device__ __forceinline__ float unmapf(unsigned u) {
  unsigned b = (u & 0x80000000u) ? (u ^ 0x80000000u) : ~u;
  return __uint_as_float(b);
}
__device__ __forceinline__ float sigmoidf(float x) {
  return 1.0f / (1.0f + __expf(-x));
}

// ---------------- 0: zero control region ----------------
__global__ void zero_kernel(unsigned* w) {
  for (int i = blockIdx.x * blockDim.x + threadIdx.x; i < ZERO_WORDS;
       i += gridDim.x * blockDim.x)
    w[i] = 0u;
}

// ---------------- 1: per-level histogram (LDS-privatized) ----------------
__global__ void hist_kernel(const float* __restrict__ cls, int n,
                            unsigned* __restrict__ histG) {
  __shared__ unsigned h[NBINS];
  for (int i = threadIdx.x; i < NBINS; i += blockDim.x) h[i] = 0u;
  __syncthreads();
  const int stride = gridDim.x * blockDim.x;
  for (int i = blockIdx.x * blockDim.x + threadIdx.x; i < n; i += stride) {
    __builtin_prefetch(cls + i + stride, 0, 1);  // global_prefetch_b8
    unsigned u = mapf(cls[i]);
    atomicAdd(&h[u >> 20], 1u);
  }
  __syncthreads();
  for (int i = threadIdx.x; i < NBINS; i += blockDim.x)
    if (h[i]) atomicAdd(&histG[i], h[i]);
}

// ---------------- 2: find threshold bin per level ----------------
__global__ void scan_kernel(const unsigned* __restrict__ hist,
                            unsigned* Tbin, unsigned* above, unsigned* rsel) {
  int t = threadIdx.x;
  if (t < N_LVL) {
    unsigned k = (t == 2) ? (unsigned)K2 : (unsigned)K0;
    const unsigned* h = hist + t * NBINS;
    unsigned cum = 0, T = 0, ab = 0, r = k;
    for (int b = NBINS - 1; b >= 0; --b) {
      unsigned c = h[b];
      if (cum + c >= k) { T = (unsigned)b; ab = cum; r = k - cum; break; }
      cum += c;
    }
    Tbin[t] = T; above[t] = ab; rsel[t] = r;
  }
}

// ---------------- 3: gather strictly-above + threshold-bin side buffer ----
__global__ void gather_kernel(const float* __restrict__ cls, int n, int level,
                              int base, const unsigned* __restrict__ Tbin,
                              unsigned* mainCnt, unsigned* sideCnt,
                              unsigned* __restrict__ candU,
                              unsigned* __restrict__ candIdx,
                              unsigned* __restrict__ sideU,
                              unsigned* __restrict__ sideIdx) {
  unsigned T = Tbin[level];
  const int stride = gridDim.x * blockDim.x;
  for (int e = blockIdx.x * blockDim.x + threadIdx.x; e < n; e += stride) {
    __builtin_prefetch(cls + e + stride, 0, 1);
    unsigned u = mapf(cls[e]);
    unsigned bin = u >> 20;
    if (bin > T) {
      unsigned p = atomicAdd(&mainCnt[level], 1u);
      candU[base + p] = u;
      candIdx[base + p] = (unsigned)e;
    } else if (bin == T) {
      unsigned p = atomicAdd(&sideCnt[level], 1u);
      if (p < SIDE_CAP) {
        sideU[level * SIDE_CAP + p] = u;
        sideIdx[level * SIDE_CAP + p] = (unsigned)e;
      }
    }
  }
}

// ---------------- bitonic sort (descending) of SORTN u64 keys in LDS ------
__device__ __forceinline__ void bitonic_desc(unsigned long long* k, int tid,
                                             int nthreads) {
  for (int len = 2; len <= SORTN; len <<= 1) {
    for (int j = len >> 1; j > 0; j >>= 1) {
      __syncthreads();
      for (int p = tid; p < SORTN / 2; p += nthreads) {
        int l = p & (j - 1);
        int i = (p << 1) - l;
        int x = i + j;
        bool up = (i & len) == 0;
        unsigned long long a = k[i], b = k[x];
        if ((a < b) == up) { k[i] = b; k[x] = a; }
      }
    }
  }
  __syncthreads();
}

// ---------------- 4: exact top-r inside threshold bin (per level) --------
__global__ __launch_bounds__(1024) void side_select_kernel(
    const unsigned* __restrict__ sideU, const unsigned* __restrict__ sideIdx,
    const unsigned* sideCnt, const unsigned* above, const unsigned* rsel,
    unsigned* __restrict__ candU, unsigned* __restrict__ candIdx) {
  __shared__ unsigned long long k[SORTN];
  int level = blockIdx.x;
  int tid = threadIdx.x;
  unsigned m = sideCnt[level];
  if (m > SIDE_CAP) m = SIDE_CAP;
  const unsigned* su = sideU + level * SIDE_CAP;
  const unsigned* si = sideIdx + level * SIDE_CAP;
  for (int t = tid; t < SORTN; t += 1024)
    k[t] = (t < (int)m)
               ? (((unsigned long long)su[t] << 32) |
                  (unsigned long long)(0xFFFFFFFFu - si[t]))  // score desc, idx asc
               : 0ull;
  __syncthreads();
  bitonic_desc(k, tid, 1024);
  unsigned r = rsel[level];
  unsigned ab = above[level];
  int base = level * K0;  // 0, 2000, 4000
  for (int t = tid; t < (int)r; t += 1024) {
    unsigned long long key = k[t];
    candU[base + ab + t] = (unsigned)(key >> 32);
    candIdx[base + ab + t] = 0xFFFFFFFFu - (unsigned)(key & 0xFFFFFFFFull);
  }
}

// ---------------- 5: build global sort keys ----------------
__global__ void build_keys_kernel(const unsigned* __restrict__ candU,
                                  const unsigned* __restrict__ candIdx,
                                  unsigned long long* __restrict__ keys) {
  int s = blockIdx.x * blockDim.x + threadIdx.x;
  if (s >= SORTN) return;
  if (s >= NCAND) { keys[s] = (unsigned long long)s; return; }
  unsigned u = candU[s];
  unsigned idx = candIdx[s];
  int level = s < K0 ? 0 : (s < (K0 + K1) ? 1 : 2);
  float score = sigmoidf(unmapf(u));
  bool valid = score > CONF_THRESH;
  unsigned vl = valid ? 3u : (2u - (unsigned)level);          // invalid: level asc
  unsigned tie16 = 0xFFFFu - ((idx >> 5) & 0xFFFFu);          // idx asc tiebreak
  keys[s] = ((unsigned long long)vl << 61) | ((unsigned long long)u << 29) |
            ((unsigned long long)tie16 << 13) | (unsigned long long)s;
}

// ---------------- 6: global sort ----------------
__global__ __launch_bounds__(1024) void sort_kernel(unsigned long long* keys) {
  __shared__ unsigned long long k[SORTN];
  int tid = threadIdx.x;
  for (int t = tid; t < SORTN; t += 1024) k[t] = keys[t];
  __syncthreads();
  bitonic_desc(k, tid, 1024);
  for (int t = tid; t < SORTN; t += 1024) keys[t] = k[t];
}

// ---------------- 7: finalize ordered outputs + offset boxes -------------
__global__ void finalize_kernel(const unsigned* __restrict__ candU,
                                const unsigned* __restrict__ candIdx,
                                const unsigned long long* __restrict__ keys,
                                const float* __restrict__ box0,
                                const float* __restrict__ box1,
                                const float* __restrict__ box2,
                                const int* __restrict__ imw,
                                const int* __restrict__ imh,
                                float* __restrict__ out,
                                float* __restrict__ sbox,
                                unsigned* __restrict__ removed) {
  int t = blockIdx.x * blockDim.x + threadIdx.x;
  if (t >= NCAND) return;
  unsigned long long key = keys[t];
  unsigned slot = (unsigned)(key & 0x1FFFull);
  int level = slot < K0 ? 0 : (slot < (K0 + K1) ? 1 : 2);
  unsigned idx = candIdx[slot];
  unsigned u = candU[slot];
  int label = (int)(idx % NUM_CLASSES);
  int anchor = (int)(idx / NUM_CLASSES);
  const float* bx = (level == 0) ? box0 : ((level == 1) ? box1 : box2);
  float b0 = bx[anchor * 4 + 0], b1 = bx[anchor * 4 + 1];
  float b2 = bx[anchor * 4 + 2], b3 = bx[anchor * 4 + 3];
  float score = sigmoidf(unmapf(u));
  float sx = 1.0f / (float)imw[0], sy = 1.0f / (float)imh[0];
  out[t * 4 + 0] = fminf(fmaxf(b0 * sx, 0.f), 1.f);
  out[t * 4 + 1] = fminf(fmaxf(b1 * sy, 0.f), 1.f);
  out[t * 4 + 2] = fminf(fmaxf(b2 * sx, 0.f), 1.f);
  out[t * 4 + 3] = fminf(fmaxf(b3 * sy, 0.f), 1.f);
  out[NCAND * 4 + t] = score;
  out[NCAND * 5 + t] = (float)label;
  float off = CLASS_OFFSET * (float)label;
  sbox[t * 4 + 0] = b0 + off;
  sbox[t * 4 + 1] = b1 + off;
  sbox[t * 4 + 2] = b2 + off;
  sbox[t * 4 + 3] = b3 + off;
  if (!(score > CONF_THRESH)) atomicOr(&removed[t >> 5], 1u << (t & 31));
}

// ---------------- 8: IoU suppression bitmask (WMMA union + VALU inter) ---
__device__ __forceinline__ float box_area4(const float* b) {
  return fmaxf(b[2] - b[0], 0.f) * fmaxf(b[3] - b[1], 0.f);
}
__device__ __forceinline__ bool iou_pred(const float* rb, const float* cb,
                                         float unionBase, int i, int j) {
  float ltx = fmaxf(rb[0], cb[0]);
  float lty = fmaxf(rb[1], cb[1]);
  float rbx = fminf(rb[2], cb[2]);
  float rby = fminf(rb[3], cb[3]);
  float iw = fmaxf(rbx - ltx, 0.f);
  float ih = fmaxf(rby - lty, 0.f);
  float inter = iw * ih;
  float uni = fmaxf(unionBase - inter, 1e-9f);  // union = area_i+area_j-inter
  // uni > 0 always  =>  inter/uni > T  <=>  inter > T*uni  (no division)
  return (inter > NMS_THRESH * uni) && (j > i);
}

__global__ __launch_bounds__(32) void mask_kernel(
    const float* __restrict__ sbox, unsigned* __restrict__ mask) {
  const int cw = blockIdx.x;   // column word 0..174
  const int rt = blockIdx.y;   // row tile 0..349
  const int R0 = rt * 16;
  const int C0 = cw * 32;
  const int lane = threadIdx.x;
  if (C0 + 31 <= R0) {         // whole word has j <= i -> zero
    if (lane < 16) mask[(size_t)(R0 + lane) * NW + cw] = 0u;
    return;
  }
  __shared__ alignas(16) float rbox[16][4];
  __shared__ alignas(16) float cbox[32][4];
#if defined(HAVE_ASYNC_LDS)
  {
    // prototype (from clang diagnostic): (int4 AS1* gsrc, int4 AS3* ldst, imm, imm)
    if (lane < 16)
      __builtin_amdgcn_global_load_async_to_lds_b128(
          (AS_GLOBAL v4i*)(sbox + (size_t)(R0 + lane) * 4),
          (AS_LDS v4i*)&rbox[lane][0], 0, 0);
    __builtin_amdgcn_global_load_async_to_lds_b128(
        (AS_GLOBAL v4i*)(sbox + (size_t)(C0 + lane) * 4),
        (AS_LDS v4i*)&cbox[lane][0], 0, 0);
    wait_asynccnt0();
  }
  __syncthreads();
#else
  if (lane < 16) {
#pragma unroll
    for (int c = 0; c < 4; ++c) rbox[lane][c] = sbox[(R0 + lane) * 4 + c];
  }
#pragma unroll
  for (int c = 0; c < 4; ++c) cbox[lane][c] = sbox[(C0 + lane) * 4 + c];
  __syncthreads();
#endif

  // A(16x4) = [area_row, 1, 0, 0]; B(4x16) = [1; area_col; 0; 0]
  // lanes 0-15 hold K=0,1 ; lanes 16-31 hold K=2,3 (zeros)
  float areaR = 0.f, areaC0 = 0.f, areaC1 = 0.f;
  if (lane < 16) {
    areaR = box_area4(rbox[lane]);
    areaC0 = box_area4(cbox[lane]);
    areaC1 = box_area4(cbox[lane + 16]);
  }
  v2f a;  a.x = (lane < 16) ? areaR : 0.f;  a.y = (lane < 16) ? 1.f : 0.f;
  v2f bA; bA.x = (lane < 16) ? 1.f : 0.f;   bA.y = (lane < 16) ? areaC0 : 0.f;
  v2f bB; bB.x = (lane < 16) ? 1.f : 0.f;   bB.y = (lane < 16) ? areaC1 : 0.f;
  v8f c0 = {}, c1 = {};
  // union additive part: c[v]@lane = area_i[R0+v+8*(lane>=16)] + area_j[col]
  c0 = __builtin_amdgcn_wmma_f32_16x16x4_f32(false, a, false, bA, (short)0, c0,
                                             false, false);
  c1 = __builtin_amdgcn_wmma_f32_16x16x4_f32(false, a, false, bB, (short)0, c1,
                                             false, false);

  const int n = lane & 15;
  const int half = lane >> 4;
  float cb0[4], cb1[4];
#pragma unroll
  for (int c = 0; c < 4; ++c) { cb0[c] = cbox[n][c]; cb1[c] = cbox[n + 16][c]; }

#pragma unroll
  for (int v = 0; v < 8; ++v) {
    int i = R0 + v + half * 8;
    float rb[4];
#pragma unroll
    for (int c = 0; c < 4; ++c) rb[c] = rbox[v + half * 8][c];
    bool p0 = iou_pred(rb, cb0, c0[v], i, C0 + n);
    bool p1 = iou_pred(rb, cb1, c1[v], i, C0 + 16 + n);
    unsigned bal0 = (unsigned)__ballot(p0);
    unsigned bal1 = (unsigned)__ballot(p1);
    unsigned rowA = (bal0 & 0xFFFFu) | ((bal1 & 0xFFFFu) << 16);  // row R0+v
    unsigned rowB = (bal0 >> 16) | ((bal1 >> 16) << 16);          // row R0+v+8
    if (lane == 0)  mask[(size_t)(R0 + v) * NW + cw] = rowA;
    if (lane == 16) mask[(size_t)(R0 + v + 8) * NW + cw] = rowB;
  }
}

// ---------------- 9: sequential NMS scan over bitmasks -------------------
__global__ __launch_bounds__(192) void nms_kernel(
    const unsigned* __restrict__ removedInit,
    const unsigned* __restrict__ mask, float* __restrict__ keepOut) {
  __shared__ unsigned rem[NW];
  int tid = threadIdx.x;
  if (tid < NW) rem[tid] = removedInit[tid];
  __syncthreads();
  for (int i = 0; i < NCAND; ++i) {
    unsigned wv = rem[i >> 5];
    bool active = ((wv >> (i & 31)) & 1u) == 0u;
    if (tid == 0) keepOut[i] = active ? 1.f : 0.f;
    __syncthreads();
    if (active) {
      const unsigned* mrow = mask + (size_t)i * NW;
      for (int w = tid; w < NW; w += 192) rem[w] |= mrow[w];
    }
    __syncthreads();
  }
}

// ---------------- launcher ----------------
extern "C" void kernel_launch(void* const* d_in, const int* in_sizes, int n_in,
                              void* d_out, int out_size, void* d_ws,
                              size_t ws_size, hipStream_t stream) {
  // setup_inputs order: cls_p3, box_p3, cls_p4, box_p4, cls_p5, box_p5, img_w, img_h
  const float* cls[3] = {(const float*)d_in[0], (const float*)d_in[2],
                         (const float*)d_in[4]};
  const float* box[3] = {(const float*)d_in[1], (const float*)d_in[3],
                         (const float*)d_in[5]};
  const int* imw = (const int*)d_in[6];
  const int* imh = (const int*)d_in[7];
  const int ncls[3] = {in_sizes[0], in_sizes[2], in_sizes[4]};
  const int base[3] = {0, K0, K0 + K1};

  char* ws = (char*)d_ws;
  unsigned* hist = (unsigned*)(ws + OFF_HIST);
  unsigned* mainCnt = (unsigned*)(ws + OFF_MAINCNT);
  unsigned* sideCnt = (unsigned*)(ws + OFF_SIDECNT);
  unsigned* Tbin = (unsigned*)(ws + OFF_TBIN);
  unsigned* above = (unsigned*)(ws + OFF_ABOVE);
  unsigned* rsel = (unsigned*)(ws + OFF_RSEL);
  unsigned* removed = (unsigned*)(ws + OFF_REMOVED);
  unsigned* candU = (unsigned*)(ws + OFF_CANDU);
  unsigned* candIdx = (unsigned*)(ws + OFF_CANDIDX);
  unsigned* sideU = (unsigned*)(ws + OFF_SIDEU);
  unsigned* sideIdx = (unsigned*)(ws + OFF_SIDEIDX);
  unsigned long long* keys = (unsigned long long*)(ws + OFF_KEYS);
  float* sbox = (float*)(ws + OFF_SBOX);
  unsigned* maskm = (unsigned*)(ws + OFF_MASK);
  float* out = (float*)d_out;

  zero_kernel<<<64, 256, 0, stream>>>((unsigned*)ws);
  for (int l = 0; l < 3; ++l)
    hist_kernel<<<1024, 256, 0, stream>>>(cls[l], ncls[l], hist + l * NBINS);
  scan_kernel<<<1, 32, 0, stream>>>(hist, Tbin, above, rsel);
  for (int l = 0; l < 3; ++l)
    gather_kernel<<<1024, 256, 0, stream>>>(cls[l], ncls[l], l, base[l], Tbin,
                                            mainCnt, sideCnt, candU, candIdx,
                                            sideU, sideIdx);
  side_select_kernel<<<3, 1024, 0, stream>>>(sideU, sideIdx, sideCnt, above,
                                             rsel, candU, candIdx);
  build_keys_kernel<<<(SORTN + 255) / 256, 256, 0, stream>>>(candU, candIdx,
                                                             keys);
  sort_kernel<<<1, 1024, 0, stream>>>(keys);
  finalize_kernel<<<(NCAND + 255) / 256, 256, 0, stream>>>(
      candU, candIdx, keys, box[0], box[1], box[2], imw, imh, out, sbox,
      removed);
  mask_kernel<<<dim3(NW, ROWTILES), 32, 0, stream>>>(sbox, maskm);
  nms_kernel<<<1, 192, 0, stream>>>(removed, maskm, out + NCAND * 6);
}